// DESMO_2860448219149
// MI455X (gfx1250) — compile-verified
//
#include <hip/hip_runtime.h>
#include <hip/hip_bf16.h>
#include <math.h>

// Problem constants (from reference): N=100000, M=1000, R=2, NUM_TERMS=6, K=12 rank
#define NN 100000
#define MM 1000
#define KK 12
#define MTILES 63     // ceil(1000/16)
#define NTILES 6250   // 100000/16 exact
#define TOTAL_WAVES (MTILES * NTILES)

typedef __attribute__((ext_vector_type(2))) float v2f;
typedef __attribute__((ext_vector_type(8))) float v8f;

// ---------------------------------------------------------------------------
// Prep: n-side factor A[n][12] + out1 (latent_spatial, N x 2)
// ---------------------------------------------------------------------------
__global__ void desmo_prep_n(const float* __restrict__ phi,
                             const float* __restrict__ pod,
                             const float* __restrict__ c_coef,
                             const float* __restrict__ omega,
                             float* __restrict__ An,
                             float* __restrict__ out1) {
    int n = blockIdx.x * 256 + threadIdx.x;
    if (n >= NN) return;
    float y0 = phi[n]      * pod[2 * n + 0];   // phi_scaled[0,n]
    float y1 = phi[NN + n] * pod[2 * n + 1];   // phi_scaled[1,n]
    out1[2 * n + 0] = y0;
    out1[2 * n + 1] = y1;
    float* a = An + (size_t)n * KK;
    a[0]  = c_coef[0];
    a[1]  = c_coef[1] * y0;
    a[2]  = c_coef[2] * y1;
    a[3]  = c_coef[3] * y0 * y0;
    a[4]  = c_coef[4] * y0 * y1;
    a[5]  = c_coef[5] * y1 * y1;
    a[6]  = sinf(omega[0] * y0);
    a[7]  = sinf(omega[3] * y1);
    a[8]  = cosf(omega[1] * y0);
    a[9]  = cosf(omega[4] * y1);
    a[10] = tanhf(omega[2] * y0);
    a[11] = tanhf(omega[5] * y1);
}

// ---------------------------------------------------------------------------
// Prep: m-side factor B[m][12] + out2 (z_values copy, 6 x M)
// ---------------------------------------------------------------------------
__global__ void desmo_prep_m(const float* __restrict__ z,
                             const float* __restrict__ zsin,
                             const float* __restrict__ zcos,
                             const float* __restrict__ ztanh,
                             const float* __restrict__ sc,
                             const float* __restrict__ cc,
                             const float* __restrict__ tc,
                             float* __restrict__ Bm,
                             float* __restrict__ out2) {
    int m = blockIdx.x * 256 + threadIdx.x;
    if (m >= MM) return;
    float* b = Bm + (size_t)m * KK;
#pragma unroll
    for (int t = 0; t < 6; ++t) {
        float zv = z[t * MM + m];
        b[t] = zv;
        out2[t * MM + m] = zv;
    }
    b[6]  = sc[0] * zsin[m];
    b[7]  = sc[1] * zsin[MM + m];
    b[8]  = cc[0] * zcos[m];
    b[9]  = cc[1] * zcos[MM + m];
    b[10] = tc[0] * ztanh[m];
    b[11] = tc[1] * ztanh[MM + m];
}

// ---------------------------------------------------------------------------
// Main: rank-12 GEMM via V_WMMA_F32_16X16X4_F32, one wave per 16x16 tile.
//   out0[m,n] = sum_k Bm[m][k] * An[n][k]
// A operand (16x4, MxK):  lane l, vgpr v -> [row=l%16][k=2*(l/16)+v]
// B operand (4x16, KxN):  lane l, vgpr v -> [k=2*(l/16)+v][col=l%16]
// C/D (16x16):            lane l, vgpr v -> [m=v+8*(l/16)][n=l%16]
// ---------------------------------------------------------------------------
__global__ __launch_bounds__(256) void desmo_wmma(const float* __restrict__ An,
                                                  const float* __restrict__ Bm,
                                                  float* __restrict__ out0) {
    int gw = (blockIdx.x * 256 + (int)threadIdx.x) >> 5;   // global wave id
    if (gw >= TOTAL_WAVES) return;                          // wave-uniform exit
    int tn = gw % NTILES;          // fast along n: consecutive waves stream rows
    int tm = gw / NTILES;
    int m0 = tm * 16;
    int n0 = tn * 16;

    int lane = (int)threadIdx.x & 31;
    int row  = lane & 15;
    int half = lane >> 4;

    // m-side rows clamp (only last m-tile has rows >= 1000; garbage is masked at store)
    int mrow = m0 + row;
    if (mrow > MM - 1) mrow = MM - 1;

    const float* ap = Bm + (size_t)mrow * KK + 2 * half;       // WMMA A operand
    const float* bp = An + (size_t)(n0 + row) * KK + 2 * half; // WMMA B operand

    v2f a0 = *(const v2f*)(ap + 0);
    v2f a1 = *(const v2f*)(ap + 4);
    v2f a2 = *(const v2f*)(ap + 8);
    v2f b0 = *(const v2f*)(bp + 0);
    v2f b1 = *(const v2f*)(bp + 4);
    v2f b2 = *(const v2f*)(bp + 8);

    v8f c = {};
    c = __builtin_amdgcn_wmma_f32_16x16x4_f32(false, a0, false, b0, (short)0, c, false, false);
    c = __builtin_amdgcn_wmma_f32_16x16x4_f32(false, a1, false, b1, (short)0, c, false, false);
    c = __builtin_amdgcn_wmma_f32_16x16x4_f32(false, a2, false, b2, (short)0, c, false, false);

    // D[v + 8*half][row] -> out0[(m0 + v + 8*half) * N + (n0 + row)]
    int    mbase = m0 + 8 * half;
    size_t col   = (size_t)(n0 + row);
    float* p     = out0 + (size_t)mbase * NN + col;

    if (tm != MTILES - 1) {
        // Common path (98.4% of waves): tile fully in-range, no exec juggling.
        // Store offsets v*NN*4 (<= 2.8MB) fit the 24-bit immediate offset.
#pragma unroll
        for (int v = 0; v < 8; ++v) {
            __builtin_nontemporal_store(c[v], p + (size_t)v * NN);
        }
    } else {
        // Edge tile: rows 992..1007, guard each row.
#pragma unroll
        for (int v = 0; v < 8; ++v) {
            if (mbase + v < MM) {
                __builtin_nontemporal_store(c[v], p + (size_t)v * NN);
            }
        }
    }
}

// ---------------------------------------------------------------------------
// Fallback (no workspace): one thread per n, loops over m. Used only if the
// harness workspace is too small for the factor matrices.
// ---------------------------------------------------------------------------
__global__ void desmo_fallback(const float* __restrict__ phi,
                               const float* __restrict__ pod,
                               const float* __restrict__ c_coef,
                               const float* __restrict__ z,
                               const float* __restrict__ zsin,
                               const float* __restrict__ zcos,
                               const float* __restrict__ ztanh,
                               const float* __restrict__ sc,
                               const float* __restrict__ cc,
                               const float* __restrict__ tc,
                               const float* __restrict__ omega,
                               float* __restrict__ out0,
                               float* __restrict__ out1) {
    int n = blockIdx.x * 256 + threadIdx.x;
    if (n >= NN) return;
    float y0 = phi[n]      * pod[2 * n + 0];
    float y1 = phi[NN + n] * pod[2 * n + 1];
    out1[2 * n + 0] = y0;
    out1[2 * n + 1] = y1;
    float a[KK];
    a[0]  = c_coef[0];
    a[1]  = c_coef[1] * y0;
    a[2]  = c_coef[2] * y1;
    a[3]  = c_coef[3] * y0 * y0;
    a[4]  = c_coef[4] * y0 * y1;
    a[5]  = c_coef[5] * y1 * y1;
    a[6]  = sinf(omega[0] * y0);
    a[7]  = sinf(omega[3] * y1);
    a[8]  = cosf(omega[1] * y0);
    a[9]  = cosf(omega[4] * y1);
    a[10] = tanhf(omega[2] * y0);
    a[11] = tanhf(omega[5] * y1);
    for (int m = 0; m < MM; ++m) {
        float acc = 0.f;
#pragma unroll
        for (int t = 0; t < 6; ++t) acc += a[t] * z[t * MM + m];
        acc += a[6]  * sc[0] * zsin[m]       + a[7]  * sc[1] * zsin[MM + m];
        acc += a[8]  * cc[0] * zcos[m]       + a[9]  * cc[1] * zcos[MM + m];
        acc += a[10] * tc[0] * ztanh[m]      + a[11] * tc[1] * ztanh[MM + m];
        out0[(size_t)m * NN + n] = acc;
    }
}

__global__ void desmo_copy_z(const float* __restrict__ z, float* __restrict__ out2) {
    int i = blockIdx.x * 256 + threadIdx.x;
    if (i < 6 * MM) out2[i] = z[i];
}

// ---------------------------------------------------------------------------
extern "C" void kernel_launch(void* const* d_in, const int* in_sizes, int n_in,
                              void* d_out, int out_size, void* d_ws, size_t ws_size,
                              hipStream_t stream) {
    // input order: X, phi, POD_modes, c_coef, z_values, zsin, zcos, ztanh,
    //              sin_coef, cos_coef, tanh_coef, omega
    const float* phi   = (const float*)d_in[1];
    const float* pod   = (const float*)d_in[2];
    const float* ccoef = (const float*)d_in[3];
    const float* z     = (const float*)d_in[4];
    const float* zsin  = (const float*)d_in[5];
    const float* zcos  = (const float*)d_in[6];
    const float* ztanh = (const float*)d_in[7];
    const float* sc    = (const float*)d_in[8];
    const float* cc    = (const float*)d_in[9];
    const float* tc    = (const float*)d_in[10];
    const float* omega = (const float*)d_in[11];

    float* out0 = (float*)d_out;                        // (M, N) = final_reconstruction.T
    float* out1 = out0 + (size_t)MM * NN;               // (N, 2) latent_spatial
    float* out2 = out1 + (size_t)NN * 2;                // (6, M) z_values copy

    size_t needA = (size_t)NN * KK * sizeof(float);     // 4.8 MB
    size_t needB = (size_t)MM * KK * sizeof(float);     // 48 KB

    if (ws_size >= needA + needB) {
        float* An = (float*)d_ws;
        float* Bm = (float*)((char*)d_ws + needA);      // needA is 16B-aligned

        desmo_prep_n<<<(NN + 255) / 256, 256, 0, stream>>>(phi, pod, ccoef, omega, An, out1);
        desmo_prep_m<<<(MM + 255) / 256, 256, 0, stream>>>(z, zsin, zcos, ztanh, sc, cc, tc, Bm, out2);

        int blocks = (TOTAL_WAVES * 32 + 255) / 256;
        desmo_wmma<<<blocks, 256, 0, stream>>>(An, Bm, out0);
    } else {
        desmo_fallback<<<(NN + 255) / 256, 256, 0, stream>>>(
            phi, pod, ccoef, z, zsin, zcos, ztanh, sc, cc, tc, omega, out0, out1);
        desmo_copy_z<<<(6 * MM + 255) / 256, 256, 0, stream>>>(z, out2);
    }
}